// PtModule_76166950027677
// MI455X (gfx1250) — compile-verified
//
#include <hip/hip_runtime.h>

// out = 2*x + 5 - 3/(1+x), elementwise over 8192*8192 fp32.
// HBM-bound: 512 MiB traffic @ 23.3 TB/s => ~23 us ideal.
// Hot kernel: branch-free full tiles, 4x b128 NT loads/stores per thread with
// immediate offsets (clause-friendly), rcp+1NR instead of precise divide.

typedef __attribute__((ext_vector_type(4))) float vfloat4;

static __device__ __forceinline__ float fused_op(float x) {
    // 2*x + 5 - 3/(1+x)
    const float t = 1.0f + x;
    // Fast reciprocal (v_rcp_f32, ~1 ulp) + one Newton-Raphson step:
    // r1 = r0 + r0*(1 - t*r0)  -> ~0.5-1 ulp of 1/t.
    float r = __builtin_amdgcn_rcpf(t);
    const float e = __builtin_fmaf(-t, r, 1.0f);
    r = __builtin_fmaf(r, e, r);
    // 2x + 5 - 3r
    return __builtin_fmaf(x, 2.0f, 5.0f) - 3.0f * r;
}

constexpr int kThreads = 256;  // 8 wave32s per block
constexpr int kILP = 4;        // 4 independent b128 loads in flight per thread

// Branch-free hot kernel: every block processes exactly kThreads*kILP float4s.
__global__ __launch_bounds__(kThreads) void fused_main_kernel(
    const float* __restrict__ in, float* __restrict__ out) {
    const vfloat4* __restrict__ in4 = (const vfloat4*)in;
    vfloat4* __restrict__ out4 = (vfloat4*)out;

    // One 64-bit address computation; per-slot displacement j*kThreads*16 B
    // (= j*4096) folds into the load/store immediate offset field.
    const long base = (long)blockIdx.x * (kThreads * kILP) + threadIdx.x;
    const vfloat4* __restrict__ pin = in4 + base;
    vfloat4* __restrict__ pout = out4 + base;

    vfloat4 v[kILP];
#pragma unroll
    for (int j = 0; j < kILP; ++j) {
        v[j] = __builtin_nontemporal_load(&pin[j * kThreads]);
    }

#pragma unroll
    for (int j = 0; j < kILP; ++j) {
        vfloat4 r;
#pragma unroll
        for (int k = 0; k < 4; ++k) {
            r[k] = fused_op(v[j][k]);
        }
        __builtin_nontemporal_store(r, &pout[j * kThreads]);
    }
}

// Generic scalar tail (empty for 8192*8192; kept for correctness on any n).
__global__ __launch_bounds__(kThreads) void fused_tail_kernel(
    const float* __restrict__ in, float* __restrict__ out, int start, int n) {
    const int i = start + blockIdx.x * blockDim.x + threadIdx.x;
    if (i < n) {
        out[i] = fused_op(in[i]);
    }
}

extern "C" void kernel_launch(void* const* d_in, const int* in_sizes, int n_in,
                              void* d_out, int out_size, void* d_ws, size_t ws_size,
                              hipStream_t stream) {
    (void)n_in; (void)d_ws; (void)ws_size; (void)out_size;
    const float* in = (const float*)d_in[0];
    float* out = (float*)d_out;
    const int n = in_sizes[0];  // 8192*8192 = 67108864

    constexpr int kElemsPerBlock = kThreads * kILP * 4;  // 4096 elements
    const int full_blocks = n / kElemsPerBlock;          // 16384 for this n
    const int done = full_blocks * kElemsPerBlock;
    const int rem = n - done;                            // 0 for this n

    if (full_blocks > 0) {
        fused_main_kernel<<<full_blocks, kThreads, 0, stream>>>(in, out);
    }
    if (rem > 0) {
        const int tail_blocks = (rem + kThreads - 1) / kThreads;
        fused_tail_kernel<<<tail_blocks, kThreads, 0, stream>>>(in, out, done, n);
    }
}